// GCNNetwork_40673340293824
// MI455X (gfx1250) — compile-verified
//
#include <hip/hip_runtime.h>
#include <hip/hip_bf16.h>

#define F_IN 512
#define HID  16
#define CLS  40

typedef __attribute__((ext_vector_type(2))) float v2f;
typedef __attribute__((ext_vector_type(8))) float v8f;

// ---------------- degree / norm ----------------

__global__ __launch_bounds__(256) void k_deg_init(float* deg, int n) {
    int i = blockIdx.x * 256 + threadIdx.x;
    if (i < n) deg[i] = 1.0f;               // self loop
}

__global__ __launch_bounds__(256) void k_deg_count(const int* __restrict__ src,
                                                   float* deg, int e) {
    int i = blockIdx.x * 256 + threadIdx.x;
    if (i < e) unsafeAtomicAdd(&deg[src[i]], 1.0f);
}

__global__ __launch_bounds__(256) void k_rsqrt(float* deg, int n) {
    int i = blockIdx.x * 256 + threadIdx.x;
    if (i < n) deg[i] = rsqrtf(deg[i]);     // in-place: deg -> dinv
}

// ---------------- GEMM1: t1 = x @ W1^T + b1  (N x 512 -> N x 16) ----------------
// One wave per 16-row tile. FP32 WMMA 16x16x4, K stepped by 4 (128 WMMAs/tile).
// A frag (16x4 f32): lanes 0-15 -> M=lane, K={k,k+1}; lanes 16-31 -> M=lane-16, K={k+2,k+3}
// B frag (4x16 f32): B[k][n] = W1[n][k], mirrored lane layout.

__global__ __launch_bounds__(256) void k_gemm1(const float* __restrict__ x,
                                               const float* __restrict__ W1,
                                               const float* __restrict__ b1,
                                               float* __restrict__ t1, int n) {
    int wid  = (blockIdx.x * 256 + threadIdx.x) >> 5;
    int lane = threadIdx.x & 31;
    int row0 = wid * 16;
    if (row0 >= n) return;                       // wave-uniform exit

    int hl = lane & 15;
    int kk = (lane >> 4) << 1;                   // 0 or 2
    int arow = row0 + hl; if (arow > n - 1) arow = n - 1;

    const float* xp = x  + (size_t)arow * F_IN + kk;
    const float* wp = W1 + (size_t)hl   * F_IN + kk;

    v8f acc = {};
    for (int k = 0; k < F_IN; k += 4) {
        v2f a = *(const v2f*)(xp + k);
        v2f b = *(const v2f*)(wp + k);
        acc = __builtin_amdgcn_wmma_f32_16x16x4_f32(false, a, false, b,
                                                    (short)0, acc, false, false);
    }

    int rbase = row0 + ((lane >> 4) << 3);       // C/D: VGPR v -> row rbase+v, col hl
    float bias = b1[hl];
#pragma unroll
    for (int v = 0; v < 8; ++v) {
        int r = rbase + v;
        if (r < n) t1[(size_t)r * HID + hl] = acc[v] + bias;
    }
}

// ---------------- aggregation layer 1 (16 features) ----------------

__global__ __launch_bounds__(256) void k_self16(const float* __restrict__ t1,
                                                const float* __restrict__ dinv,
                                                float* __restrict__ agg, int total) {
    int i = blockIdx.x * 256 + threadIdx.x;
    if (i >= total) return;
    int node = i >> 4;
    float d = dinv[node];
    agg[i] = d * d * t1[i];                      // self-loop term: dinv[i]^2 * t1[i]
}

__global__ __launch_bounds__(256) void k_edge16(const int* __restrict__ src,
                                                const int* __restrict__ dst,
                                                const float* __restrict__ t1,
                                                const float* __restrict__ dinv,
                                                float* __restrict__ agg, int total) {
    int i = blockIdx.x * 256 + threadIdx.x;
    if (i >= total) return;
    int e = i >> 4, f = i & 15;
    int s = src[e], d = dst[e];
    float v = dinv[s] * dinv[d] * t1[(size_t)s * HID + f];
    unsafeAtomicAdd(&agg[(size_t)d * HID + f], v);
}

// ---------------- GEMM2: t2 = relu(agg1) @ W2^T + b2  (N x 16 -> N x 40) ----------------
// One wave per 16-row tile; 3 N-tiles (cols 0-15,16-31,32-39), 4 K-steps each.

__global__ __launch_bounds__(256) void k_gemm2(const float* __restrict__ agg1,
                                               const float* __restrict__ W2,
                                               const float* __restrict__ b2,
                                               float* __restrict__ t2, int n) {
    int wid  = (blockIdx.x * 256 + threadIdx.x) >> 5;
    int lane = threadIdx.x & 31;
    int row0 = wid * 16;
    if (row0 >= n) return;

    int hl = lane & 15;
    int kk = (lane >> 4) << 1;
    int arow = row0 + hl; if (arow > n - 1) arow = n - 1;

    float a_row[HID];
    const float* ap = agg1 + (size_t)arow * HID;
#pragma unroll
    for (int k = 0; k < HID; ++k) a_row[k] = fmaxf(ap[k], 0.0f);   // relu

    int rbase = row0 + ((lane >> 4) << 3);

    for (int n0 = 0; n0 < CLS; n0 += 16) {
        int col = n0 + hl;
        v8f acc = {};
#pragma unroll
        for (int ks = 0; ks < HID; ks += 4) {
            v2f a; a.x = a_row[ks + kk]; a.y = a_row[ks + kk + 1];
            v2f b;
            if (col < CLS) {
                b.x = W2[col * HID + ks + kk];
                b.y = W2[col * HID + ks + kk + 1];
            } else { b.x = 0.0f; b.y = 0.0f; }
            acc = __builtin_amdgcn_wmma_f32_16x16x4_f32(false, a, false, b,
                                                        (short)0, acc, false, false);
        }
        if (col < CLS) {
            float bias = b2[col];
#pragma unroll
            for (int v = 0; v < 8; ++v) {
                int r = rbase + v;
                if (r < n) t2[(size_t)r * CLS + col] = acc[v] + bias;
            }
        }
    }
}

// ---------------- aggregation layer 2 (40 features) ----------------

__global__ __launch_bounds__(256) void k_self40(const float* __restrict__ t2,
                                                const float* __restrict__ dinv,
                                                float* __restrict__ agg, int total) {
    int i = blockIdx.x * 256 + threadIdx.x;
    if (i >= total) return;
    int node = i / CLS;
    float d = dinv[node];
    agg[i] = d * d * t2[i];
}

__global__ __launch_bounds__(256) void k_edge40(const int* __restrict__ src,
                                                const int* __restrict__ dst,
                                                const float* __restrict__ t2,
                                                const float* __restrict__ dinv,
                                                float* __restrict__ agg, int total) {
    int i = blockIdx.x * 256 + threadIdx.x;
    if (i >= total) return;
    int e = i / CLS, c = i - e * CLS;
    int s = src[e], d = dst[e];
    float v = dinv[s] * dinv[d] * t2[(size_t)s * CLS + c];
    unsafeAtomicAdd(&agg[(size_t)d * CLS + c], v);
}

// ---------------- log_softmax over 40 classes, in-place on d_out ----------------

__global__ __launch_bounds__(256) void k_logsoftmax(float* __restrict__ out, int n) {
    int i = blockIdx.x * 256 + threadIdx.x;
    if (i >= n) return;
    float* r = out + (size_t)i * CLS;
    float mx = r[0];
#pragma unroll 8
    for (int c = 1; c < CLS; ++c) mx = fmaxf(mx, r[c]);
    float s = 0.0f;
#pragma unroll 8
    for (int c = 0; c < CLS; ++c) s += expf(r[c] - mx);
    float ls = logf(s);
#pragma unroll 8
    for (int c = 0; c < CLS; ++c) r[c] = r[c] - mx - ls;
}

// ---------------- host launch ----------------

extern "C" void kernel_launch(void* const* d_in, const int* in_sizes, int n_in,
                              void* d_out, int out_size, void* d_ws, size_t ws_size,
                              hipStream_t stream) {
    const float* x  = (const float*)d_in[0];
    const int*   ei = (const int*)d_in[1];      // [2, E] flattened
    const float* W1 = (const float*)d_in[2];
    const float* b1 = (const float*)d_in[3];
    const float* W2 = (const float*)d_in[4];
    const float* b2 = (const float*)d_in[5];
    float* out = (float*)d_out;

    const int N = out_size / CLS;
    const int E = in_sizes[1] / 2;
    const int* src = ei;
    const int* dst = ei + E;

    // workspace layout (floats): deg/dinv [N] | t1 [N*16] | agg1 [N*16] | t2 [N*40]
    float* ws   = (float*)d_ws;
    float* deg  = ws;                            // becomes dinv in-place
    float* t1   = deg  + (size_t)N;
    float* agg1 = t1   + (size_t)N * HID;
    float* t2   = agg1 + (size_t)N * HID;

    const int tiles = (N + 15) / 16;

    // 1) degrees & dinv
    k_deg_init <<<(N + 255) / 256, 256, 0, stream>>>(deg, N);
    k_deg_count<<<(E + 255) / 256, 256, 0, stream>>>(src, deg, E);
    k_rsqrt    <<<(N + 255) / 256, 256, 0, stream>>>(deg, N);

    // 2) t1 = x @ W1^T + b1   (WMMA f32)
    k_gemm1<<<(tiles + 7) / 8, 256, 0, stream>>>(x, W1, b1, t1, N);

    // 3) agg1 = normalized scatter-add of t1 (self-loop init + edge atomics)
    {
        int tot = N * HID;
        k_self16<<<(tot + 255) / 256, 256, 0, stream>>>(t1, deg, agg1, tot);
        int etot = E * HID;
        k_edge16<<<(etot + 255) / 256, 256, 0, stream>>>(src, dst, t1, deg, agg1, etot);
    }

    // 4) t2 = relu(agg1) @ W2^T + b2   (WMMA f32)
    k_gemm2<<<(tiles + 7) / 8, 256, 0, stream>>>(agg1, W2, b2, t2, N);

    // 5) agg2 directly into d_out
    {
        int tot = N * CLS;
        k_self40<<<(tot + 255) / 256, 256, 0, stream>>>(t2, deg, out, tot);
        int etot = E * CLS;
        k_edge40<<<(etot + 255) / 256, 256, 0, stream>>>(src, dst, t2, deg, out, etot);
    }

    // 6) log_softmax in place
    k_logsoftmax<<<(N + 255) / 256, 256, 0, stream>>>(out, N);
}